// MultiHeadAttentionQuantum_5368709120611
// MI455X (gfx1250) — compile-verified
//
#include <hip/hip_runtime.h>
#include <math.h>

typedef float v2f __attribute__((ext_vector_type(2)));
typedef float v8f __attribute__((ext_vector_type(8)));

#define BB   16
#define SS   1024
#define EE   8
#define DKK  4

// Raw v_exp_f32 (no denorm fixup): our exponents are bounded (|x| < 3).
#if __has_builtin(__builtin_amdgcn_exp2f)
#define FAST_EXP2(x) __builtin_amdgcn_exp2f(x)
#else
#define FAST_EXP2(x) exp2f(x)
#endif

// Closed-form qlayer:
//   a0=in0+t0, a1=in1+t1, a2=in2+in3+t2, a3=t3, a4..7=in4..7+t4..7
//   ck = cos(ak);  out0 = c1*c2*...*c7 ;  out_w = c0*c1*...*cw  (w>=1)
__global__ void qlayer_kernel(const float* __restrict__ in,
                              const float* __restrict__ theta,
                              float* __restrict__ out, int ntok)
{
    int t = blockIdx.x * blockDim.x + threadIdx.x;
    if (t >= ntok) return;
    const float* p = in + (size_t)t * 8;
    float4 f0 = *(const float4*)(p);
    float4 f1 = *(const float4*)(p + 4);

    float a0 = f0.x + theta[0];
    float a1 = f0.y + theta[1];
    float a2 = f0.z + f0.w + theta[2];
    float a3 = theta[3];
    float a4 = f1.x + theta[4];
    float a5 = f1.y + theta[5];
    float a6 = f1.z + theta[6];
    float a7 = f1.w + theta[7];

    float c0 = __cosf(a0), c1 = __cosf(a1), c2 = __cosf(a2), c3 = __cosf(a3);
    float c4 = __cosf(a4), c5 = __cosf(a5), c6 = __cosf(a6), c7 = __cosf(a7);

    float o1 = c0 * c1;
    float o2 = o1 * c2;
    float o3 = o2 * c3;
    float o4 = o3 * c4;
    float o5 = o4 * c5;
    float o6 = o5 * c6;
    float o7 = o6 * c7;
    float o0 = ((c1 * c2) * (c3 * c4)) * ((c5 * c6) * c7);

    float* q = out + (size_t)t * 8;
    *(float4*)(q)     = make_float4(o0, o1, o2, o3);
    *(float4*)(q + 4) = make_float4(o4, o5, o6, o7);
}

// XOR-butterfly add across the 16 lanes of each half-wave using DPP ROW_XMASK
// (0x160 | mask): pure VALU, no LDS traffic, no dscnt waits.
template<int MASK>
__device__ __forceinline__ float dpp_xor_add(float v)
{
    int o = __builtin_amdgcn_update_dpp(0, __float_as_int(v),
                                        0x160 | MASK, 0xf, 0xf, true);
    return v + __int_as_float(o);
}

__device__ __forceinline__ float half_reduce_add(float v)
{
    v = dpp_xor_add<1>(v);
    v = dpp_xor_add<2>(v);
    v = dpp_xor_add<4>(v);
    v = dpp_xor_add<8>(v);
    return v;
}

// One wave (32 threads) per 16-query tile of one (b,h) problem.
// Score tile via V_WMMA_F32_16X16X4_F32 (dk == K == 4), streamed over 64 key
// tiles. |h| <= 1 elementwise => |score| <= 4, so softmax needs no max
// tracking: the key loop is load + wmma + exp + fma only (no cross-lane ops).
__global__ void attn_kernel(const float* __restrict__ qkv,   // [B,S,8]
                            float* __restrict__ concat)      // [B,S,8]
{
    const int lane = threadIdx.x;        // 0..31, full wave
    const int qt   = blockIdx.x;         // 0..63  (query tile)
    const int bh   = blockIdx.y;         // 0..31
    const int b    = bh >> 1;
    const int h    = bh & 1;

    const float* Hbase = qkv + (size_t)b * SS * EE + h * DKK;  // row s: Hbase + s*8
    const int  col = lane & 15;
    const bool hi  = lane >= 16;

    // A operand: 16 queries x 4 (lane%16 = M row; lane-half picks K {0,1}/{2,3})
    const int q0 = qt * 16;
    float4 qv = *(const float4*)(Hbase + (size_t)(q0 + col) * 8);
    v2f A;
    A.x = hi ? qv.z : qv.x;
    A.y = hi ? qv.w : qv.y;

    float l[8], acc[8][4];
#pragma unroll
    for (int r = 0; r < 8; ++r) {
        l[r] = 0.0f;
        acc[r][0] = acc[r][1] = acc[r][2] = acc[r][3] = 0.0f;
    }

    // exp(score/2) = exp2(score * 0.5 * log2(e))
    const float K2 = 0.5f * 1.4426950408889634f;

#pragma unroll 2
    for (int kt = 0; kt < 64; ++kt) {
        float4 kv = *(const float4*)(Hbase + (size_t)(kt * 16 + col) * 8);
        v2f Bop;
        Bop.x = hi ? kv.z : kv.x;
        Bop.y = hi ? kv.w : kv.y;

        v8f cz = {0.f, 0.f, 0.f, 0.f, 0.f, 0.f, 0.f, 0.f};
        // D = A(16x4) * B(4x16) + 0  -> 16x16 score tile (rows=queries, cols=keys)
        v8f s = __builtin_amdgcn_wmma_f32_16x16x4_f32(
            false, A, false, Bop, (short)0, cz, false, false);

#pragma unroll
        for (int r = 0; r < 8; ++r) {
            float e = FAST_EXP2(s[r] * K2);      // bounded: |arg| <= 2.89
            l[r] += e;
            acc[r][0] = fmaf(e, kv.x, acc[r][0]);
            acc[r][1] = fmaf(e, kv.y, acc[r][1]);
            acc[r][2] = fmaf(e, kv.z, acc[r][2]);
            acc[r][3] = fmaf(e, kv.w, acc[r][3]);
        }
    }

    // Reduce partials over the 16 lanes of each half (key columns).
#pragma unroll
    for (int r = 0; r < 8; ++r) {
        l[r]      = half_reduce_add(l[r]);
        acc[r][0] = half_reduce_add(acc[r][0]);
        acc[r][1] = half_reduce_add(acc[r][1]);
        acc[r][2] = half_reduce_add(acc[r][2]);
        acc[r][3] = half_reduce_add(acc[r][3]);
    }

    if (col == 0) {                      // lanes 0 and 16 write 8 rows each
        const int rbase = hi ? 8 : 0;
        float* obase = concat + (size_t)b * SS * EE + h * DKK;
#pragma unroll
        for (int r = 0; r < 8; ++r) {
            float inv = 1.0f / l[r];
            int qq = q0 + rbase + r;
            float4 o = make_float4(acc[r][0] * inv, acc[r][1] * inv,
                                   acc[r][2] * inv, acc[r][3] * inv);
            *(float4*)(obase + (size_t)qq * 8) = o;
        }
    }
}

extern "C" void kernel_launch(void* const* d_in, const int* in_sizes, int n_in,
                              void* d_out, int out_size, void* d_ws, size_t ws_size,
                              hipStream_t stream)
{
    const float* x     = (const float*)d_in[0];   // (16,1024,8)
    const float* theta = (const float*)d_in[1];   // (8,)
    float* out = (float*)d_out;                   // (16,1024,8)

    const int ntok = BB * SS;                     // 16384
    float* qkv    = (float*)d_ws;                 // 131072 floats (512 KB)
    float* concat = (float*)d_ws + (size_t)ntok * EE;

    dim3 qblk(256), qgrd((ntok + 255) / 256);
    qlayer_kernel<<<qgrd, qblk, 0, stream>>>(x, theta, qkv, ntok);

    attn_kernel<<<dim3(SS / 16, BB * 2), 32, 0, stream>>>(qkv, concat);

    qlayer_kernel<<<qgrd, qblk, 0, stream>>>(concat, theta, out, ntok);
}